// DenseGCN_30880814858419
// MI455X (gfx1250) — compile-verified
//
#include <hip/hip_runtime.h>
#include <hip/hip_bf16.h>

typedef __attribute__((ext_vector_type(2))) float v2f;
typedef __attribute__((ext_vector_type(8))) float v8f;

#define TM 5  // row-tiles per wave: 50000 = 625 * (5*16)

// ---------------------------------------------------------------------------
// out[m, n] = relu( (A1[m,:] * rowscale[m]) @ W1[:, n] + bias[n]
//                   (+ A2[m,:] @ W2[:, n] if W2 != null) )
// W1/W2 are [K, Fout] row-major. One wave computes an 80x16 output stripe
// using V_WMMA_F32_16X16X4_F32. EXEC is all-ones for every wave that reaches
// the WMMA (early-out is wave-uniform).
// ---------------------------------------------------------------------------
__global__ void sage_gemm_wmma(const float* __restrict__ A1, int lda1,
                               const float* __restrict__ rowscale,
                               const float* __restrict__ W1,
                               const float* __restrict__ bias,
                               const float* __restrict__ A2, int lda2,
                               const float* __restrict__ W2,
                               float* __restrict__ out, int ldo,
                               int M, int K, int Fout) {
    const int lane   = threadIdx.x & 31;
    const int wid    = (int)((blockIdx.x * blockDim.x + threadIdx.x) >> 5);
    const int tilesN = Fout >> 4;
    const int tn     = wid % tilesN;
    const int tmBase = (wid / tilesN) * (16 * TM);
    if (tmBase >= M) return;  // wave-uniform

    const int ml = lane & 15;          // row within tile for A, col within tile for B/C
    const int kk = (lane >> 4) << 1;   // 0 (lanes 0-15) or 2 (lanes 16-31)
    const int n  = tn * 16 + ml;

    v8f acc[TM];
    const v8f zero8 = {};
#pragma unroll
    for (int t = 0; t < TM; ++t) acc[t] = zero8;

    float rs[TM];
#pragma unroll
    for (int t = 0; t < TM; ++t)
        rs[t] = rowscale ? rowscale[tmBase + t * 16 + ml] : 1.0f;

    // ---- first term: (A1 * rowscale) @ W1 ----
    for (int k = 0; k < K; k += 4) {
        v2f b;
        b.x = W1[(size_t)(k + kk) * Fout + n];
        b.y = W1[(size_t)(k + kk + 1) * Fout + n];
#pragma unroll
        for (int t = 0; t < TM; ++t) {
            const float* ar = A1 + (size_t)(tmBase + t * 16 + ml) * lda1 + k + kk;
            v2f a;
            a.x = ar[0] * rs[t];
            a.y = ar[1] * rs[t];
            acc[t] = __builtin_amdgcn_wmma_f32_16x16x4_f32(
                false, a, false, b, (short)0, acc[t], false, false);
        }
    }

    // ---- second term: A2 @ W2 (root/self weights) ----
    if (W2 != nullptr) {
        for (int k = 0; k < K; k += 4) {
            v2f b;
            b.x = W2[(size_t)(k + kk) * Fout + n];
            b.y = W2[(size_t)(k + kk + 1) * Fout + n];
#pragma unroll
            for (int t = 0; t < TM; ++t) {
                const float* ar = A2 + (size_t)(tmBase + t * 16 + ml) * lda2 + k + kk;
                v2f a;
                a.x = ar[0];
                a.y = ar[1];
                acc[t] = __builtin_amdgcn_wmma_f32_16x16x4_f32(
                    false, a, false, b, (short)0, acc[t], false, false);
            }
        }
    }

    // ---- bias + relu + store ----
    const float bn     = bias[n];
    const int   rhalf  = (lane >> 4) << 3;  // 0 or 8
#pragma unroll
    for (int t = 0; t < TM; ++t) {
#pragma unroll
        for (int j = 0; j < 8; ++j) {
            const int row = tmBase + t * 16 + rhalf + j;
            float v = acc[t][j] + bn;
            out[(size_t)row * ldo + tn * 16 + ml] = fmaxf(v, 0.0f);
        }
    }
}

// ---------------------------------------------------------------------------
// Edge scatter: agg[dst, :] += h[src, :]  (one thread = one edge x 4 floats)
// Coalesced float4 gathers, native fp32 atomics into L2.
// ---------------------------------------------------------------------------
__global__ void scatter_add_kernel(const float* __restrict__ h, int ldh,
                                   const int* __restrict__ src,
                                   const int* __restrict__ dst,
                                   float* __restrict__ agg, int lda,
                                   int F4, int E) {
    const long long t     = (long long)blockIdx.x * blockDim.x + threadIdx.x;
    const long long total = (long long)E * F4;
    if (t >= total) return;
    const int e = (int)(t / F4);
    const int c = (int)(t % F4);
    const int s = src[e];
    const int d = dst[e];
    const float4 v = *(const float4*)(h + (size_t)s * ldh + (size_t)c * 4);
    float* p = agg + (size_t)d * lda + (size_t)c * 4;
    __hip_atomic_fetch_add(p + 0, v.x, __ATOMIC_RELAXED, __HIP_MEMORY_SCOPE_AGENT);
    __hip_atomic_fetch_add(p + 1, v.y, __ATOMIC_RELAXED, __HIP_MEMORY_SCOPE_AGENT);
    __hip_atomic_fetch_add(p + 2, v.z, __ATOMIC_RELAXED, __HIP_MEMORY_SCOPE_AGENT);
    __hip_atomic_fetch_add(p + 3, v.w, __ATOMIC_RELAXED, __HIP_MEMORY_SCOPE_AGENT);
}

__global__ void degree_kernel(const int* __restrict__ dst,
                              float* __restrict__ deg, int E) {
    const int e = blockIdx.x * blockDim.x + threadIdx.x;
    if (e < E)
        __hip_atomic_fetch_add(&deg[dst[e]], 1.0f, __ATOMIC_RELAXED,
                               __HIP_MEMORY_SCOPE_AGENT);
}

__global__ void invdeg_kernel(const float* __restrict__ deg,
                              float* __restrict__ invdeg, int N) {
    const int i = blockIdx.x * blockDim.x + threadIdx.x;
    if (i < N) invdeg[i] = 1.0f / fmaxf(deg[i], 1.0f);
}

__global__ void zero_kernel(float4* __restrict__ p, int n4) {
    const int i = blockIdx.x * blockDim.x + threadIdx.x;
    if (i < n4) p[i] = make_float4(0.f, 0.f, 0.f, 0.f);
}

// ---------------------------------------------------------------------------
extern "C" void kernel_launch(void* const* d_in, const int* in_sizes, int n_in,
                              void* d_out, int out_size, void* d_ws, size_t ws_size,
                              hipStream_t stream) {
    const float* x   = (const float*)d_in[0];
    const int*   ei  = (const int*)d_in[1];   // [2, E] (jax default int32)
    const float* Wp  = (const float*)d_in[2];
    const float* bp  = (const float*)d_in[3];
    const float* Wl1 = (const float*)d_in[4];
    const float* bl1 = (const float*)d_in[5];
    const float* Wr1 = (const float*)d_in[6];
    const float* Wl2 = (const float*)d_in[7];
    const float* bl2 = (const float*)d_in[8];
    const float* Wr2 = (const float*)d_in[9];
    const float* Wl3 = (const float*)d_in[10];
    const float* bl3 = (const float*)d_in[11];
    const float* Wr3 = (const float*)d_in[12];

    const int N  = 50000;
    const int IN = 256, F1 = 128, F2 = 128, F3 = 256;
    const int E  = in_sizes[1] / 2;
    const int* srcIdx = ei;
    const int* dstIdx = ei + E;

    // Workspace layout (floats): cat[N,384] | agg[N,384] | deg[N] | invdeg[N]
    float* ws     = (float*)d_ws;
    float* cat    = ws;                           // x_p | h1 | h2 columns
    float* agg    = cat + (size_t)N * 384;
    float* deg    = agg + (size_t)N * 384;
    float* invdeg = deg + N;

    const int TB = 256;
    auto gemm = [&](const float* A1, int lda1, const float* rsc,
                    const float* W1, const float* bias,
                    const float* A2, int lda2, const float* W2,
                    float* out, int ldo, int K, int Fout) {
        const int tilesN = Fout / 16;
        const int waves  = (N / (16 * TM)) * tilesN;   // exact: N % 80 == 0
        const int blocks = (waves * 32 + TB - 1) / TB;
        sage_gemm_wmma<<<blocks, TB, 0, stream>>>(A1, lda1, rsc, W1, bias,
                                                  A2, lda2, W2, out, ldo,
                                                  N, K, Fout);
    };
    auto zero = [&](float* p, long long nflt) {
        const int n4 = (int)(nflt / 4);
        zero_kernel<<<(n4 + TB - 1) / TB, TB, 0, stream>>>((float4*)p, n4);
    };
    auto scatter = [&](const float* h, int ldh, int F, float* a, int lda) {
        const int F4 = F / 4;
        const long long total = (long long)E * F4;
        const unsigned blocks = (unsigned)((total + TB - 1) / TB);
        scatter_add_kernel<<<blocks, TB, 0, stream>>>(h, ldh, srcIdx, dstIdx,
                                                      a, lda, F4, E);
    };

    // degrees -> invdeg (shared by all three layers)
    zero(deg, N);
    degree_kernel<<<(E + TB - 1) / TB, TB, 0, stream>>>(dstIdx, deg, E);
    invdeg_kernel<<<(N + TB - 1) / TB, TB, 0, stream>>>(deg, invdeg, N);

    // x_p = relu(x @ Wp + bp) -> cat[:, 0:128]
    gemm(x, IN, nullptr, Wp, bp, nullptr, 0, nullptr, cat + 0, 384, IN, F1);

    // layer 1: aggregate x, h1 -> cat[:, 128:256]
    zero(agg, (long long)N * 256);
    scatter(x, IN, IN, agg, 256);
    gemm(agg, 256, invdeg, Wl1, bl1, x, IN, Wr1, cat + 128, 384, IN, F1);

    // layer 2: input = cat[:, 0:256] (ld=384), h2 -> cat[:, 256:384]
    zero(agg, (long long)N * 256);
    scatter(cat, 384, 256, agg, 256);
    gemm(agg, 256, invdeg, Wl2, bl2, cat, 384, Wr2, cat + 256, 384, 2 * F1, F2);

    // layer 3: input = cat[:, 0:384], h3 -> d_out [N, 256]
    zero(agg, (long long)N * 384);
    scatter(cat, 384, 384, agg, 384);
    gemm(agg, 384, invdeg, Wl3, bl3, cat, 384, Wr3, (float*)d_out, F3,
         2 * F1 + F2, F3);
}